// CDLoss_25039659336370
// MI455X (gfx1250) — compile-verified
//
#include <hip/hip_runtime.h>

typedef __attribute__((ext_vector_type(2))) float v2f;
typedef __attribute__((ext_vector_type(8))) float v8f;

// Directed chamfer: for each query row i, min_j ( |q_i|^2 + |r_j|^2 - 2 q_i.r_j ),
// summed over rows and atomically accumulated (scaled) into *out.
// gridDim = (N/128, B, 2); blockDim = 256 (8 waves, 16 query rows per wave).
__global__ __launch_bounds__(256) void chamfer_wmma_kernel(
    const float* __restrict__ srcs, const float* __restrict__ tgts,
    float* __restrict__ out, int N, float scale)
{
    const int lane = threadIdx.x & 31;
    const int wave = threadIdx.x >> 5;
    const int b    = blockIdx.y;
    const int dir  = blockIdx.z;

    // dir 0: per-src min over tgts (axis=2 of P); dir 1: per-tgt min over srcs (axis=1).
    const float* Q = (dir == 0) ? srcs : tgts;
    const float* R = (dir == 0) ? tgts : srcs;
    Q += (size_t)b * 3 * N;   // layout [B, D, N]: coord d of point n at d*N + n
    R += (size_t)b * 3 * N;

    const int rowBase = blockIdx.x * 128 + wave * 16;  // 16 query rows for this wave
    const int m  = lane & 15;
    const bool hi = (lane >= 16);

    // ---- A fragment (16x4 f32): row M = lane&15.
    // VGPR0 holds K=0 (lanes 0-15) / K=2 (lanes 16-31); VGPR1 holds K=1 / K=3.
    // A row = [qx, qy, qz, 1]
    const int qi = rowBase + m;
    const float qx = Q[qi], qy = Q[N + qi], qz = Q[2 * N + qi];
    v2f av;
    av.x = hi ? qz   : qx;
    av.y = hi ? 1.0f : qy;

    // |q|^2 sum for the 8 rows this half-wave owns (rows r+8*hi, r=0..7);
    // identical across all 16 lanes of the half (broadcast loads hit cache).
    float qnsum = 0.0f;
    #pragma unroll
    for (int r = 0; r < 8; ++r) {
        const int mm = rowBase + r + (hi ? 8 : 0);
        const float x = Q[mm], y = Q[N + mm], z = Q[2 * N + mm];
        qnsum += x * x + y * y + z * z;
    }

    // Running elementwise min over all ref tiles, in C-fragment layout.
    v8f vmin;
    #pragma unroll
    for (int r = 0; r < 8; ++r) vmin[r] = 3.0e38f;

    const int ntiles = N >> 4;
    for (int t = 0; t < ntiles; ++t) {
        // ---- B fragment (4x16 f32): col N = lane&15, same K-per-VGPR split as A.
        // B col = [-2rx, -2ry, -2rz, |r|^2]
        const int ri = t * 16 + m;
        const float rx = R[ri], ry = R[N + ri], rz = R[2 * N + ri];
        const float rr = rx * rx + ry * ry + rz * rz;
        v2f bv;
        bv.x = hi ? (-2.0f * rz) : (-2.0f * rx);
        bv.y = hi ? rr           : (-2.0f * ry);

        v8f c = {};  // fresh accumulator each tile
        // D = A*B + C :  C[i][j] = |r_j|^2 - 2 q_i . r_j
        c = __builtin_amdgcn_wmma_f32_16x16x4_f32(
                /*neg_a=*/false, av, /*neg_b=*/false, bv,
                /*c_mod=*/(short)0, c, /*reuse_a=*/false, /*reuse_b=*/false);

        #pragma unroll
        for (int r = 0; r < 8; ++r) vmin[r] = fminf(vmin[r], c[r]);
    }

    // Min-reduce across the 16 lanes of each half (cols 0-15 of the tiles).
    // xor masks 1,2,4,8 never cross the 16-lane boundary on wave32.
    float rowsum = 0.0f;
    #pragma unroll
    for (int r = 0; r < 8; ++r) {
        float v = vmin[r];
        #pragma unroll
        for (int d = 1; d < 16; d <<= 1)
            v = fminf(v, __shfl_xor(v, d, 32));
        rowsum += v;  // min for row r + 8*hi (uniform across the half now)
    }

    // halfsum: sum over this half's 8 rows of (|q_i|^2 + min_j(...))
    const float halfsum = rowsum + qnsum;
    const float other   = __shfl_xor(halfsum, 16, 32);
    if (lane == 0) {
        atomicAdd(out, (halfsum + other) * scale);  // global_atomic_add_f32
    }
}

extern "C" void kernel_launch(void* const* d_in, const int* in_sizes, int n_in,
                              void* d_out, int out_size, void* d_ws, size_t ws_size,
                              hipStream_t stream) {
    (void)in_sizes; (void)n_in; (void)out_size; (void)d_ws; (void)ws_size;
    const float* srcs = (const float*)d_in[0];
    const float* tgts = (const float*)d_in[1];
    float* out = (float*)d_out;

    const int B = 8, N = 4096;
    const float scale = 1.0f / (float)(B * N);  // each directed term is a mean over B*N

    hipMemsetAsync(out, 0, sizeof(float), stream);

    dim3 grid(N / 128, B, 2);   // 32 blocks x 8 batches x 2 directions
    chamfer_wmma_kernel<<<grid, 256, 0, stream>>>(srcs, tgts, out, N, scale);
}